// SAGE_46170898432376
// MI455X (gfx1250) — compile-verified
//
#include <hip/hip_runtime.h>

#define NNODES 50000
#define NEDGES 600000
#define FEAT   128

typedef __attribute__((ext_vector_type(2))) float v2f;
typedef __attribute__((ext_vector_type(8))) float v8f;

// ---------------------------------------------------------------- utilities
__global__ void zero_f32_vec4(float4* __restrict__ p, int n4) {
  int i = blockIdx.x * blockDim.x + threadIdx.x;
  if (i < n4) p[i] = make_float4(0.f, 0.f, 0.f, 0.f);
}

__global__ void deg_kernel(const int* __restrict__ dst, float* __restrict__ deg, int E) {
  int e = blockIdx.x * blockDim.x + threadIdx.x;
  if (e < E) atomicAdd(&deg[dst[e]], 1.0f);
}

__global__ void invdeg_kernel(float* __restrict__ deg, int n) {
  int i = blockIdx.x * blockDim.x + threadIdx.x;
  if (i < n) deg[i] = 1.0f / fmaxf(deg[i], 1.0f);
}

// Wt[c][k] = W[k][c]  (W: [FEAT rows, cols]); consecutive tid -> consecutive k
// so stores are fully coalesced; strided reads are fine for a 64KB one-shot.
__global__ void transpose_w(const float* __restrict__ W, float* __restrict__ Wt,
                            int cols, int total) {
  int idx = blockIdx.x * blockDim.x + threadIdx.x;
  if (idx >= total) return;
  int c = idx >> 7;            // / FEAT
  int k = idx & (FEAT - 1);    // % FEAT
  Wt[idx] = W[k * cols + c];
}

// ------------------------------------------------- edge scatter (mean numerator)
// one thread = one float4 chunk of one edge; 32 chunks cover F=128.
// consecutive lanes -> consecutive chunks of same edge => coalesced loads and
// contiguous f32 atomic adds into the destination row (L2-resident, 192MB L2).
__global__ void scatter_kernel(const float* __restrict__ h,
                               const int* __restrict__ src,
                               const int* __restrict__ dst,
                               float* __restrict__ msum, int E) {
  unsigned tid = blockIdx.x * blockDim.x + threadIdx.x;
  int chunk = tid & 31;          // FEAT/4 = 32 chunks
  int e = tid >> 5;
  if (e >= E) return;
  int s = src[e];
  int d = dst[e];
  const float4 v = reinterpret_cast<const float4*>(h + (long)s * FEAT)[chunk];
  float* p = msum + (long)d * FEAT + chunk * 4;
  atomicAdd(p + 0, v.x);
  atomicAdd(p + 1, v.y);
  atomicAdd(p + 2, v.z);
  atomicAdd(p + 3, v.w);
}

// ------------------------------------------------- fused SAGE layer GEMM (WMMA f32)
// out[16-row strip, wave's 16-col tile] =
//    act( h @ Wself + (msum * invdeg_row) @ Wneigh + bias )
// Weights are pre-transposed ([FOUT][FEAT]) so every operand fetch — A and B —
// is one contiguous global_load_b64 of a K-pair. Branchless inner loop; the
// 47-col layer clamps the tail column's B index (garbage in an unstored D
// column is harmless) and guards only the final store.
template <int FOUT, bool RELU>
__global__ void sage_gemm(const float* __restrict__ h,       // [N, FEAT]
                          const float* __restrict__ msum,    // [N, FEAT]
                          const float* __restrict__ invdeg,  // [N]
                          const float* __restrict__ WselfT,  // [FOUT, FEAT]
                          const float* __restrict__ WneighT, // [FOUT, FEAT]
                          const float* __restrict__ bias,    // [FOUT]
                          float* __restrict__ out) {         // [N, FOUT]
  const int wave    = threadIdx.x >> 5;
  const int lane    = threadIdx.x & 31;
  const int rowBase = blockIdx.x * 16;          // N % 16 == 0 (50000/16 = 3125)
  const int colBase = wave * 16;

  // A-operand: lane holds row (lane&15), K-pair base (lane>>4)*2
  const int mh = lane & 15;
  const int kb = (lane >> 4) << 1;
  const float* hrow = h    + (long)(rowBase + mh) * FEAT + kb;
  const float* mrow = msum + (long)(rowBase + mh) * FEAT + kb;
  const float sc    = invdeg[rowBase + mh];

  // B-operand: lane holds col (lane&15), same K-pair base. Clamp tail col.
  const int col  = colBase + (lane & 15);
  const int colC = (FOUT % 16 == 0) ? col : (col < FOUT ? col : FOUT - 1);
  const float* wsp = WselfT  + (long)colC * FEAT + kb;
  const float* wnp = WneighT + (long)colC * FEAT + kb;

  v8f acc = {};
#pragma unroll
  for (int k = 0; k < FEAT; k += 4) {
    // --- self term ---
    const v2f a = *reinterpret_cast<const v2f*>(hrow + k);   // b64
    const v2f b = *reinterpret_cast<const v2f*>(wsp + k);    // b64
    acc = __builtin_amdgcn_wmma_f32_16x16x4_f32(false, a, false, b,
                                                (short)0, acc, false, false);
    // --- neighbor (mean) term: fold per-row 1/deg into A ---
    v2f a2 = *reinterpret_cast<const v2f*>(mrow + k);        // b64
    a2.x *= sc;
    a2.y *= sc;
    const v2f b2 = *reinterpret_cast<const v2f*>(wnp + k);   // b64
    acc = __builtin_amdgcn_wmma_f32_16x16x4_f32(false, a2, false, b2,
                                                (short)0, acc, false, false);
  }

  const float bv = bias[colC];
#pragma unroll
  for (int i = 0; i < 8; ++i) {
    // D layout: VGPR i, lanes 0-15 -> m=i ; lanes 16-31 -> m=i+8 ; n = lane&15
    const int m = i + ((lane >> 4) << 3);
    float v = acc[i] + bv;
    if (RELU) v = fmaxf(v, 0.0f);
    if (FOUT % 16 == 0) {
      out[(long)(rowBase + m) * FOUT + col] = v;
    } else if (col < FOUT) {
      out[(long)(rowBase + m) * FOUT + col] = v;
    }
  }
}

// ---------------------------------------------------------------- driver
extern "C" void kernel_launch(void* const* d_in, const int* in_sizes, int n_in,
                              void* d_out, int out_size, void* d_ws, size_t ws_size,
                              hipStream_t stream) {
  const float* x       = (const float*)d_in[0];
  const int*   src     = (const int*)d_in[1];
  const int*   dst     = (const int*)d_in[2];
  const float* W_self0 = (const float*)d_in[3];
  const float* W_ngh0  = (const float*)d_in[4];
  const float* b0      = (const float*)d_in[5];
  const float* W_self1 = (const float*)d_in[6];
  const float* W_ngh1  = (const float*)d_in[7];
  const float* b1      = (const float*)d_in[8];
  const float* W_self2 = (const float*)d_in[9];
  const float* W_ngh2  = (const float*)d_in[10];
  const float* b2      = (const float*)d_in[11];
  float* out = (float*)d_out;

  // workspace layout (256B aligned slices)
  char* ws = (char*)d_ws;
  size_t off = 0;
  auto take = [&](size_t bytes) {
    char* p = ws + off;
    off += (bytes + 255) & ~size_t(255);
    return p;
  };
  float* deg  = (float*)take((size_t)NNODES * sizeof(float));            // deg -> invdeg
  float* msum = (float*)take((size_t)NNODES * FEAT * sizeof(float));
  float* h1   = (float*)take((size_t)NNODES * FEAT * sizeof(float));
  float* h2   = (float*)take((size_t)NNODES * FEAT * sizeof(float));
  float* Ws0T = (float*)take((size_t)FEAT * FEAT * sizeof(float));
  float* Wn0T = (float*)take((size_t)FEAT * FEAT * sizeof(float));
  float* Ws1T = (float*)take((size_t)FEAT * FEAT * sizeof(float));
  float* Wn1T = (float*)take((size_t)FEAT * FEAT * sizeof(float));
  float* Ws2T = (float*)take((size_t)47 * FEAT * sizeof(float));
  float* Wn2T = (float*)take((size_t)47 * FEAT * sizeof(float));
  (void)ws_size;

  const int T = 256;
  const int n4Deg   = (NNODES + 3) / 4;
  const int n4Msum  = (NNODES * FEAT) / 4;
  const int gZeroDeg  = (n4Deg + T - 1) / T;
  const int gZeroMsum = (n4Msum + T - 1) / T;
  const int gNodes    = (NNODES + T - 1) / T;
  const int gEdges    = (NEDGES + T - 1) / T;
  const int gScatter  = (NEDGES * 32 + T - 1) / T;   // 32 chunks per edge
  const int gRows     = NNODES / 16;                 // 3125 row strips
  const int tW128     = FEAT * FEAT;                 // 16384
  const int tW47      = 47 * FEAT;                   // 6016
  const int gW128     = (tW128 + T - 1) / T;
  const int gW47      = (tW47 + T - 1) / T;

  // weight transposes (one-shot, L2-resident afterwards)
  transpose_w<<<gW128, T, 0, stream>>>(W_self0, Ws0T, 128, tW128);
  transpose_w<<<gW128, T, 0, stream>>>(W_ngh0,  Wn0T, 128, tW128);
  transpose_w<<<gW128, T, 0, stream>>>(W_self1, Ws1T, 128, tW128);
  transpose_w<<<gW128, T, 0, stream>>>(W_ngh1,  Wn1T, 128, tW128);
  transpose_w<<<gW47,  T, 0, stream>>>(W_self2, Ws2T, 47,  tW47);
  transpose_w<<<gW47,  T, 0, stream>>>(W_ngh2,  Wn2T, 47,  tW47);

  // degrees -> inverse degrees (shared by all 3 layers)
  zero_f32_vec4<<<gZeroDeg, T, 0, stream>>>((float4*)deg, n4Deg);
  deg_kernel<<<gEdges, T, 0, stream>>>(dst, deg, NEDGES);
  invdeg_kernel<<<gNodes, T, 0, stream>>>(deg, NNODES);

  // ---- layer 0: x -> h1 (ReLU) ----
  zero_f32_vec4<<<gZeroMsum, T, 0, stream>>>((float4*)msum, n4Msum);
  scatter_kernel<<<gScatter, T, 0, stream>>>(x, src, dst, msum, NEDGES);
  sage_gemm<128, true><<<gRows, 8 * 32, 0, stream>>>(x, msum, deg,
                                                     Ws0T, Wn0T, b0, h1);

  // ---- layer 1: h1 -> h2 (ReLU) ----
  zero_f32_vec4<<<gZeroMsum, T, 0, stream>>>((float4*)msum, n4Msum);
  scatter_kernel<<<gScatter, T, 0, stream>>>(h1, src, dst, msum, NEDGES);
  sage_gemm<128, true><<<gRows, 8 * 32, 0, stream>>>(h1, msum, deg,
                                                     Ws1T, Wn1T, b1, h2);

  // ---- layer 2: h2 -> out [N,47] (no act), 3 col tiles (48 >= 47) ----
  zero_f32_vec4<<<gZeroMsum, T, 0, stream>>>((float4*)msum, n4Msum);
  scatter_kernel<<<gScatter, T, 0, stream>>>(h2, src, dst, msum, NEDGES);
  sage_gemm<47, false><<<gRows, 3 * 32, 0, stream>>>(h2, msum, deg,
                                                     Ws2T, Wn2T, b2, out);

  (void)in_sizes; (void)n_in; (void)out_size;
}